// LaguerreEncoding_7670811591193
// MI455X (gfx1250) — compile-verified
//
#include <hip/hip_runtime.h>

typedef float v2f __attribute__((ext_vector_type(2)));
typedef float v8f __attribute__((ext_vector_type(8)));

#define KK     8
#define WROW   129   // HIDDEN + 1
#define HALFC  64

// Raw hardware transcendentals: v_exp_f32 (exp2) and v_log_f32 (log2).
__device__ __forceinline__ float hw_exp2(float x) { return __builtin_amdgcn_exp2f(x); }
__device__ __forceinline__ float hw_log2(float x) { return __builtin_amdgcn_logf(x); }

// ---------------------------------------------------------------------------
// Stage 0: per-k scalar folding.
//   base[k] = b[k] + W[k,0]
//   s12[k]  = sum_{i=1..128} W[k,i]
// ---------------------------------------------------------------------------
__global__ void lag_prep(const float* __restrict__ W, const float* __restrict__ b,
                         float* __restrict__ base, float* __restrict__ s12) {
  int k = threadIdx.x;
  if (k < KK) {
    const float* w = W + k * WROW;
    float s = 0.0f;
#pragma unroll
    for (int i = 1; i < WROW; ++i) s += w[i];
    base[k] = b[k] + w[0];
    s12[k]  = s;
  }
}

// ---------------------------------------------------------------------------
// Stage 1: main kernel. One wave32 handles a 16-row n-tile, loops k = 0..7.
// Dot products over the 64 softplus values are accumulated on the matrix
// pipe via V_WMMA_F32_16X16X4_F32 (A = W2 chunk replicated across rows,
// B = per-n x values), leaving VALU/trans slots free for the exp2/log2 work
// that bounds this kernel. All 16 A-operand chunks for a given k are
// preloaded into VGPRs before the WMMA loop so no per-step load waits
// interrupt the trans pipeline.
// ---------------------------------------------------------------------------
__global__ __launch_bounds__(256) void lag_main(
    const float* __restrict__ E, const float* __restrict__ W,
    const float* __restrict__ alphap, const float* __restrict__ base,
    const float* __restrict__ s12, float* __restrict__ out, int n_total) {
  const int lane = threadIdx.x & 31;
  const int wave = blockIdx.x * (blockDim.x >> 5) + (threadIdx.x >> 5);
  const int n0   = wave << 4;
  if (n0 >= n_total) return;          // wave-uniform: EXEC stays all-ones

  const int row = lane & 15;          // n within tile
  const int h   = lane >> 4;          // half-wave selects j-subcolumns

  int ne = n0 + row;
  if (ne >= n_total) ne = n_total - 1;          // clamp (keeps EXEC full)
  const float e     = E[ne];
  const float alpha = alphap[0];

  const float LOG2E = 1.4426950408889634f;
  const float LN2   = 0.6931471805599453f;

  float p = 1.0f;
  for (int k = 0; k < KK; ++k) {
    p *= e;                                     // p = e^(k+1)
    const float* __restrict__ W2 = W + k * WROW + 1 + HALFC;

    // Preload all A-operand chunks for this k: lane's pair at j-columns
    // 4s+2h, 4s+2h+1 for s = 0..15 (16 b64 loads issued back-to-back).
    v2f aw[16];
#pragma unroll
    for (int s = 0; s < 16; ++s) {
      const int cb = 4 * s + 2 * h;
      aw[s][0] = W2[cb];
      aw[s][1] = W2[cb + 1];
    }

    v8f c = {0.f, 0.f, 0.f, 0.f, 0.f, 0.f, 0.f, 0.f};
    float wsum = 0.0f;

#if __has_builtin(__builtin_amdgcn_wmma_f32_16x16x4_f32)
#pragma unroll
    for (int s = 0; s < 16; ++s) {
      const int cb = 4 * s + 2 * h;             // j-column base for this lane
      // softplus via log2/exp2: l2 = log2(1 + e^t), x = l2*ln2,
      // exp(-alpha*x) = 2^(-alpha*l2)
      float t0 = p * (float)(cb + 1);
      float t1 = p * (float)(cb + 2);
      float l0 = hw_log2(1.0f + hw_exp2(t0 * LOG2E));
      float l1 = hw_log2(1.0f + hw_exp2(t1 * LOG2E));
      wsum += hw_exp2(-alpha * l0) + hw_exp2(-alpha * l1);
      v2f bb;
      bb[0] = l0 * LN2;
      bb[1] = l1 * LN2;
      c = __builtin_amdgcn_wmma_f32_16x16x4_f32(false, aw[s], false, bb,
                                                (short)0, c, false, false);
    }
    // All 16 A-rows identical => every C element of column (lane%16) holds
    // dot(W2, x_n) for n = n0 + lane%16; c[0] is valid in every lane.
    float dot = c[0];
#else
    float dacc = 0.0f;
#pragma unroll
    for (int s = 0; s < 16; ++s) {
      const int cb = 4 * s + 2 * h;
      float t0 = p * (float)(cb + 1);
      float t1 = p * (float)(cb + 2);
      float l0 = hw_log2(1.0f + hw_exp2(t0 * LOG2E));
      float l1 = hw_log2(1.0f + hw_exp2(t1 * LOG2E));
      wsum += hw_exp2(-alpha * l0) + hw_exp2(-alpha * l1);
      dacc += aw[s][0] * (l0 * LN2) + aw[s][1] * (l1 * LN2);
    }
    float dot = dacc + __shfl_xor(dacc, 16, 32);
#endif

    // Combine the two half-wave partial sums of exp(-alpha*x_j).
    float wo    = __shfl_xor(wsum, 16, 32);
    float wmean = (wsum + wo) * (1.0f / 64.0f);

    float result = base[k] + wmean * (s12[k] - dot);

    if (lane < 16 && (n0 + row) < n_total)
      out[(size_t)k * (size_t)n_total + (size_t)(n0 + row)] = result;
  }
}

// ---------------------------------------------------------------------------
extern "C" void kernel_launch(void* const* d_in, const int* in_sizes, int n_in,
                              void* d_out, int out_size, void* d_ws, size_t ws_size,
                              hipStream_t stream) {
  const float* e     = (const float*)d_in[0];
  const float* W     = (const float*)d_in[1];
  const float* b     = (const float*)d_in[2];
  const float* alpha = (const float*)d_in[3];
  float*       out   = (float*)d_out;
  const int    n     = in_sizes[0];

  float* base = (float*)d_ws;       // 8 floats
  float* s12  = base + 8;           // 8 floats

  lag_prep<<<1, 32, 0, stream>>>(W, b, base, s12);

  const int tiles          = (n + 15) / 16;      // one wave32 per 16-row tile
  const int waves_per_blk  = 256 / 32;
  const int blocks         = (tiles + waves_per_blk - 1) / waves_per_blk;
  lag_main<<<blocks, 256, 0, stream>>>(e, W, alpha, base, s12, out, n);
}